// SampleExtractor_53841710023040
// MI455X (gfx1250) — compile-verified
//
#include <hip/hip_runtime.h>

typedef float v2f __attribute__((ext_vector_type(2)));
typedef float v8f __attribute__((ext_vector_type(8)));

#define NB    8
#define NPTS  16384
#define NCH   64
#define PROP  16

// One workgroup per batch. 1024 threads = 32 waves (wave32).
// Each wave processes groups of 16 points; the 64-channel L1 reduction is done
// on the matrix unit: D = |x-s| (16x4) x ones(4x16), accumulated over 16 k-chunks.
__global__ __launch_bounds__(1024, 1)
void fps_kernel(const float* __restrict__ src,
                float* __restrict__ outSamples,
                float* __restrict__ dminAll) {
    const int b    = blockIdx.x;
    const int t    = threadIdx.x;
    const int lane = t & 31;
    const int wave = t >> 5;

    const float* sb   = src + (size_t)b * NPTS * NCH;
    float*       dmin = dminAll + (size_t)b * NPTS;   // scratch inside sources-out region

    __shared__ float S[NCH];                 // current sample (or centroid)
    __shared__ float part[16][NCH];          // centroid partials (deterministic reduce)
    __shared__ unsigned long long bestKey;   // (dist_bits<<32) | ~idx  -> ds_max_u64
    __shared__ int bestIdx;

    // ---- centroid = mean over N (deterministic two-stage reduction) ----
    {
        const int c = t & 63, chunk = t >> 6;          // 16 chunks x 1024 pts
        float acc = 0.0f;
        const float* p = sb + (size_t)chunk * 1024 * NCH + c;
        for (int n = 0; n < 1024; ++n) acc += p[n * NCH];
        part[chunk][c] = acc;
        if (t == 0) bestKey = 0ull;
        __syncthreads();
        if (t < NCH) {
            float s = 0.0f;
            #pragma unroll
            for (int k = 0; k < 16; ++k) s += part[k][t];
            S[t] = s * (1.0f / (float)NPTS);
        }
    }

    // Static lane roles from the 16x4 f32 A layout and 16x16 f32 D layout.
    const v2f  ones  = {1.0f, 1.0f};
    const int  chOff = (lane >> 4) << 1;          // lanes 0-15: K={0,1}; 16-31: K={2,3}
    const int  m     = lane & 15;                 // A-matrix row (point within group)
    const bool writer = (lane & 8) == 0;          // lanes 0-7 own rows 0-7; 16-23 own 8-15
    const int  vsel  = lane & 7;
    const int  mrow  = vsel + ((lane >> 4) << 3);

    for (int it = 0; it < PROP; ++it) {
        __syncthreads();   // S ready, bestKey reset

        for (int g = wave; g < NPTS / 16; g += 32) {          // uniform per wave, EXEC all-1s
            const float* pp = sb + (size_t)(g * 16 + m) * NCH + chOff;
            v8f acc = {};
            #pragma unroll
            for (int kc = 0; kc < 16; ++kc) {
                float x0 = pp[kc * 4];
                float x1 = pp[kc * 4 + 1];
                float s0 = S[kc * 4 + chOff];
                float s1 = S[kc * 4 + chOff + 1];
                v2f a;
                a.x = fabsf(x0 - s0);
                a.y = fabsf(x1 - s1);
                // D += A x ones : every D column = row sums of A
                acc = __builtin_amdgcn_wmma_f32_16x16x4_f32(
                        false, a, false, ones, (short)0, acc, false, false);
            }
            if (writer) {
                float d = (vsel == 0) ? acc[0] : (vsel == 1) ? acc[1] :
                          (vsel == 2) ? acc[2] : (vsel == 3) ? acc[3] :
                          (vsel == 4) ? acc[4] : (vsel == 5) ? acc[5] :
                          (vsel == 6) ? acc[6] : acc[7];
                int pidx = g * 16 + mrow;
                // centroid round (it==0) and first-sample round (it==1) both reset dmin
                float nd = (it <= 1) ? d : fminf(dmin[pidx], d);
                dmin[pidx] = nd;   // only ever touched by this one thread -> no sync needed
                unsigned long long key =
                    ((unsigned long long)__float_as_uint(nd) << 32) |
                    (unsigned int)(~pidx);                     // ties -> lowest index wins
                atomicMax(&bestKey, key);                      // ds_max_u64 on LDS
            }
        }
        __syncthreads();   // all atomics visible

        if (t == 0) {
            bestIdx = (int)(~(unsigned int)bestKey);
            bestKey = 0ull;
        }
        __syncthreads();   // idx ready, key reset

        if (t < NCH) {
            float v = sb[(size_t)bestIdx * NCH + t];
            S[t] = v;
            outSamples[((size_t)b * PROP + it) * NCH + t] = v;
        }
    }
}

// Streams the 32 MB sources block to the second output region (overwrites the
// dmin scratch). Runs after fps_kernel on the same stream.
__global__ void copy_src_kernel(const float4* __restrict__ in,
                                float4* __restrict__ out, int n4) {
    int i = blockIdx.x * blockDim.x + threadIdx.x;
    int stride = gridDim.x * blockDim.x;
    for (; i < n4; i += stride) out[i] = in[i];
}

extern "C" void kernel_launch(void* const* d_in, const int* in_sizes, int n_in,
                              void* d_out, int out_size, void* d_ws, size_t ws_size,
                              hipStream_t stream) {
    (void)in_sizes; (void)n_in; (void)out_size; (void)d_ws; (void)ws_size;
    const float* src = (const float*)d_in[0];
    float* out        = (float*)d_out;
    float* outSamples = out;                                   // [8,16,64]
    float* outSrc     = out + (size_t)NB * PROP * NCH;         // [8,16384,64]

    fps_kernel<<<dim3(NB), dim3(1024), 0, stream>>>(src, outSamples, outSrc);

    const int n4 = (NB * NPTS * NCH) / 4;                      // 2,097,152 float4
    copy_src_kernel<<<dim3(2048), dim3(256), 0, stream>>>(
        (const float4*)src, (float4*)outSrc, n4);
}